// attention_embedding_10161892623133
// MI455X (gfx1250) — compile-verified
//
#include <hip/hip_runtime.h>

// MI455X / gfx1250, wave32. Bandwidth-bound fused kernel:
//   pass 1: stream rel once; WMMA f32 16x16x4 does the j-contraction,
//           VALU accumulates global sum of squares of (input + rel).
//   pass 2: deterministic reduction of per-block squares (1 wave).
//   pass 3: apply global scale + per-row L2 normalize in-place on d_out.

typedef __attribute__((ext_vector_type(2))) float v2f;
typedef __attribute__((ext_vector_type(8))) float v8f;

#define B_ 4
#define L_ 512
#define D_ 128
#define EPS_ 1e-12f
#define NROW (B_ * L_)

// ---------------- Kernel 1: per-(b,i) row reduction ----------------
// One block (8 waves) per row. Wave w owns j in [w*64, w*64+64).
// WMMA mapping (per wave, per (jb, dt)):
//   A (16x4, broadcast over M): A[m,k] = att[j0+k]
//     layout: lanes 0-15 -> VGPR0=K0, VGPR1=K1; lanes 16-31 -> K2, K3
//   B (4x16): B[k,n] = input[d0+n] + rel[j0+k, d0+n]   (assumed same half-split)
//   C (16x16 f32): all M rows equal = partial out[d0..d0+15]
__global__ __launch_bounds__(256) void k1_reduce(
    const float* __restrict__ att,   // [B,L,L]
    const float* __restrict__ inp,   // [B,L,D]
    const float* __restrict__ rel,   // [B,L,L,D]
    float* __restrict__ out_uns,     // [B,L,D] un-normalized accumulation
    float* __restrict__ block_sq)    // [NROW] per-row sum of squares
{
  __shared__ float s_att[L_];
  __shared__ float s_in[D_];
  __shared__ float s_part[8][D_];
  __shared__ float s_sq[8];

  const int row  = blockIdx.x;        // b*L + i
  const int t    = threadIdx.x;
  const int lane = t & 31;
  const int wave = t >> 5;
  const int half = lane >> 4;         // 0: lanes 0-15, 1: lanes 16-31
  const int n    = lane & 15;

  const float* att_row = att + (size_t)row * L_;
  const float* in_row  = inp + (size_t)row * D_;
  const float* rel_row = rel + (size_t)row * (L_ * D_);

  // Stage att row (2 KB) and input row (512 B) in LDS.
  s_att[t]       = att_row[t];
  s_att[t + 256] = att_row[t + 256];
  if (t < D_) s_in[t] = in_row[t];
  __syncthreads();

  // Per-lane copy of input[d0 + n] for each of the 8 d-chunks.
  float ind[8];
#pragma unroll
  for (int dt = 0; dt < 8; ++dt) ind[dt] = s_in[dt * 16 + n];

  v8f c[8];
  const v8f zero = {};
#pragma unroll
  for (int dt = 0; dt < 8; ++dt) c[dt] = zero;

  float sq = 0.0f;

  const int jwave = wave * 64;
  // Per-lane base pointer: row jwave + 2*half, column n.
  const float* pw = rel_row + (size_t)(jwave + half * 2) * D_ + n;

  for (int jb = 0; jb < 16; ++jb) {
    const int j0 = jwave + jb * 4;
    v2f av;
    av.x = s_att[j0 + half * 2];
    av.y = s_att[j0 + half * 2 + 1];

    const float* p = pw + (size_t)jb * 4 * D_;
    // Prefetch the next 4-row tile of this wave's stream.
    __builtin_prefetch(p + 4 * D_, 0, 0);

#pragma unroll
    for (int dt = 0; dt < 8; ++dt) {
      const float bx = p[dt * 16]      + ind[dt];  // row j0 + 2*half
      const float by = p[dt * 16 + D_] + ind[dt];  // row j0 + 2*half + 1
      v2f bv;
      bv.x = bx;
      bv.y = by;
      // c[dt] += A(att) x B(value); f32 accumulate on the matrix pipe.
      c[dt] = __builtin_amdgcn_wmma_f32_16x16x4_f32(
          false, av, false, bv, (short)0, c[dt], false, false);
      // Global-norm accumulation: each (j,d) element appears exactly once.
      sq = __builtin_fmaf(bx, bx, sq);
      sq = __builtin_fmaf(by, by, sq);
    }
  }

  // C rows are identical (broadcast A): VGPR0 of C holds partial[d0 + n]
  // for lanes 0-15 (M=0). Write this wave's 128 partials to LDS.
  if (lane < 16) {
#pragma unroll
    for (int dt = 0; dt < 8; ++dt) s_part[wave][dt * 16 + n] = c[dt][0];
  }

  // Deterministic wave reduction of sq (fixed shuffle tree).
#pragma unroll
  for (int off = 16; off > 0; off >>= 1) sq += __shfl_down(sq, off, 32);
  if (lane == 0) s_sq[wave] = sq;
  __syncthreads();

  // Fixed-order cross-wave sums.
  if (t < D_) {
    float a = 0.0f;
#pragma unroll
    for (int w = 0; w < 8; ++w) a += s_part[w][t];
    out_uns[(size_t)row * D_ + t] = a;
  }
  if (t == 0) {
    float bs = 0.0f;
#pragma unroll
    for (int w = 0; w < 8; ++w) bs += s_sq[w];
    block_sq[row] = bs;
  }
}

// ---------------- Kernel 2: deterministic global sum of squares ----------------
__global__ void k2_sumsq(const float* __restrict__ block_sq,
                         float* __restrict__ sq_total)
{
  const int lane = threadIdx.x;  // 32 threads, 1 wave
  float s = 0.0f;
  for (int i = lane; i < NROW; i += 32) s += block_sq[i];
#pragma unroll
  for (int off = 16; off > 0; off >>= 1) s += __shfl_down(s, off, 32);
  if (lane == 0) sq_total[0] = s;
}

// ---------------- Kernel 3: global scale + per-row L2 normalize ----------------
__global__ __launch_bounds__(128) void k3_finalize(
    float* __restrict__ out, const float* __restrict__ sq_total)
{
  __shared__ float s_r[D_];
  const int row = blockIdx.x;
  const int t   = threadIdx.x;

  const float g = rsqrtf(fmaxf(sq_total[0], EPS_));
  const float r = out[(size_t)row * D_ + t] * g;

  s_r[t] = r * r;
  __syncthreads();
#pragma unroll
  for (int off = 64; off > 0; off >>= 1) {
    if (t < off) s_r[t] += s_r[t + off];
    __syncthreads();
  }
  const float nrm = rsqrtf(fmaxf(s_r[0], EPS_));
  out[(size_t)row * D_ + t] = r * nrm;
}

// ---------------- Launcher ----------------
extern "C" void kernel_launch(void* const* d_in, const int* in_sizes, int n_in,
                              void* d_out, int out_size, void* d_ws, size_t ws_size,
                              hipStream_t stream) {
  (void)in_sizes; (void)n_in; (void)out_size; (void)ws_size;

  const float* att = (const float*)d_in[0];  // [B,L,L]
  const float* inp = (const float*)d_in[1];  // [B,L,D]
  const float* rel = (const float*)d_in[2];  // [B,L,L,D]
  float* out = (float*)d_out;                // [B,L,D]

  float* ws       = (float*)d_ws;
  float* sq_total = ws;                      // 1 float
  float* block_sq = ws + 1;                  // NROW floats (all written by k1)

  k1_reduce<<<NROW, 256, 0, stream>>>(att, inp, rel, out, block_sq);
  k2_sumsq<<<1, 32, 0, stream>>>(block_sq, sq_total);
  k3_finalize<<<NROW, D_, 0, stream>>>(out, sq_total);
}